// LipschitzMargin_23751169147046
// MI455X (gfx1250) — compile-verified
//
#include <hip/hip_runtime.h>
#include <math.h>

typedef __attribute__((ext_vector_type(2))) float v2f;
typedef __attribute__((ext_vector_type(8))) float v8f;

#define B_   1024
#define C_   1000
#define H_   512
#define CP_  1008           // 63 tiles * 16, padded N
#define TN_  63
#define TM_  64
#define NTILES (TM_ * TN_)  // 4032, divisible by 8 waves/block

// ---------------- kernel 1: per-row max + argmax of y ----------------
__global__ void k_rowmax(const float* __restrict__ y,
                         float* __restrict__ yj, int* __restrict__ jarr) {
  int wave = threadIdx.x >> 5, lane = threadIdx.x & 31;
  int b = blockIdx.x * 8 + wave;
  if (b >= B_) return;
  const float* row = y + (size_t)b * C_;
  float bv = -__builtin_inff(); int bi = 0;
  for (int c = lane; c < C_; c += 32) {
    float v = row[c];
    if (v > bv) { bv = v; bi = c; }
  }
#pragma unroll
  for (int off = 16; off >= 1; off >>= 1) {
    float ov = __shfl_xor(bv, off, 32);
    int   oi = __shfl_xor(bi, off, 32);
    if (ov > bv || (ov == bv && oi < bi)) { bv = ov; bi = oi; }
  }
  if (lane == 0) { yj[b] = bv; jarr[b] = bi; }
}

// ---------------- kernel 2: row norms ||w[c]||^2 ----------------
__global__ void k_norms(const float* __restrict__ w, float* __restrict__ n) {
  int wave = threadIdx.x >> 5, lane = threadIdx.x & 31;
  int c = blockIdx.x * 8 + wave;
  if (c >= C_) return;
  const float* row = w + (size_t)c * H_;
  float s = 0.f;
#pragma unroll
  for (int it = 0; it < H_ / 128; ++it) {
    float4 v = *(const float4*)(row + it * 128 + lane * 4);
    s += v.x * v.x + v.y * v.y + v.z * v.z + v.w * v.w;
  }
#pragma unroll
  for (int off = 16; off >= 1; off >>= 1) s += __shfl_xor(s, off, 32);
  if (lane == 0) n[c] = s;
}

// ---------------- kernel 3: gather argmax norms, init ybot ----------------
__global__ void k_prep(const int* __restrict__ jarr, const float* __restrict__ n,
                       float* __restrict__ nj, float* __restrict__ ybot) {
  int b = blockIdx.x * blockDim.x + threadIdx.x;
  if (b < B_) { nj[b] = n[jarr[b]]; ybot[b] = -__builtin_inff(); }
}

// ---------------- kernel 4: WMMA f32 GEMM (G = Wj * W^T) + fused epilogue ----
__global__ void __launch_bounds__(256) k_gemm(
    const float* __restrict__ y, const float* __restrict__ w,
    const int* __restrict__ jarr, const float* __restrict__ yjv,
    const float* __restrict__ n, const float* __restrict__ nj,
    const float* __restrict__ eps_p, const float* __restrict__ lip_p,
    float* __restrict__ ybot) {
  int wave = threadIdx.x >> 5, lane = threadIdx.x & 31;
  int tile = blockIdx.x * 8 + wave;          // grid sized so tile < NTILES always
  int tm = tile / TN_, tn = tile - tm * TN_;
  int half = lane >> 4;                      // ISA: lane half selects K {0,1} vs {2,3}
  int l    = lane & 15;

  // A-matrix row M=l -> gathered row w[j[tm*16+l]]; per-step load is 2 contiguous K
  int jb = jarr[tm * 16 + l];
  const float* arow = w + (size_t)jb * H_ + 2 * half;
  // B-matrix col N=l -> c = tn*16+l; B[k][n] = w[c][k] (contiguous along H)
  int c = tn * 16 + l;
  int csafe = (c < C_) ? c : 0;              // padded cols masked in epilogue
  const float* brow = w + (size_t)csafe * H_ + 2 * half;

  v8f acc = (v8f)(0.0f);
#pragma unroll 4
  for (int k = 0; k < H_; k += 4) {
    v2f av = *(const v2f*)(arow + k);
    v2f bv = *(const v2f*)(brow + k);
    // D = A(16x4) * B(4x16) + C  ->  v_wmma_f32_16x16x4_f32
    acc = __builtin_amdgcn_wmma_f32_16x16x4_f32(
        /*neg_a=*/false, av, /*neg_b=*/false, bv,
        /*c_mod=*/(short)0, acc, /*reuse_a=*/false, /*reuse_b=*/false);
  }

  float eps = *eps_p, lip = *lip_p;
  float l2 = lip * lip;
  float ncv = (c < C_) ? n[c] : 0.f;
  const float NEG_INF = -__builtin_inff();

  // C/D layout: lanes 0-15 hold (M=r, N=l); lanes 16-31 hold (M=8+r, N=l)
#pragma unroll
  for (int r = 0; r < 8; ++r) {
    int m = r + 8 * half;
    int b = tm * 16 + m;
    float val = NEG_INF;
    if (c < C_) {
      float g  = acc[r];
      float sq = l2 * (nj[b] + ncv - 2.0f * g);
      float K  = sq > 0.f ? __builtin_sqrtf(sq) : 0.f;
      float yv = y[(size_t)b * C_ + c];
      val = (yv == yjv[b]) ? NEG_INF : fmaf(eps, K, yv);
    }
    // row-max across the 16 lanes of this half (xor<16 stays within a half)
#pragma unroll
    for (int off = 1; off < 16; off <<= 1)
      val = fmaxf(val, __shfl_xor(val, off, 32));
    if (l == 0) atomicMax(&ybot[b], val);    // global_atomic_max_num_f32
  }
}

// ---------------- kernel 5: out = [y | ybot] ----------------
__global__ void k_out(const float* __restrict__ y, const float* __restrict__ ybot,
                      float* __restrict__ out) {
  int idx = blockIdx.x * blockDim.x + threadIdx.x;
  if (idx >= B_ * (C_ + 1)) return;
  int b = idx / (C_ + 1);
  int c = idx - b * (C_ + 1);
  out[idx] = (c < C_) ? y[(size_t)b * C_ + c] : ybot[b];
}

extern "C" void kernel_launch(void* const* d_in, const int* in_sizes, int n_in,
                              void* d_out, int out_size, void* d_ws, size_t ws_size,
                              hipStream_t stream) {
  const float* y   = (const float*)d_in[0];
  const float* w   = (const float*)d_in[1];
  const float* eps = (const float*)d_in[2];
  const float* lip = (const float*)d_in[3];
  float* out = (float*)d_out;

  float* wsf  = (float*)d_ws;
  int*   jarr = (int*)wsf;                 // B ints
  float* yj   = wsf + B_;                  // B
  float* n    = wsf + 2 * B_;              // CP_
  float* nj   = wsf + 2 * B_ + CP_;        // B
  float* ybot = wsf + 3 * B_ + CP_;        // B   (~20 KB total workspace)

  k_rowmax<<<B_ / 8, 256, 0, stream>>>(y, yj, jarr);
  k_norms<<<(C_ + 7) / 8, 256, 0, stream>>>(w, n);
  k_prep<<<(B_ + 255) / 256, 256, 0, stream>>>(jarr, n, nj, ybot);
  k_gemm<<<NTILES / 8, 256, 0, stream>>>(y, w, jarr, yj, n, nj, eps, lip, ybot);
  k_out<<<(B_ * (C_ + 1) + 255) / 256, 256, 0, stream>>>(y, ybot, out);
}